// SSIM_21749714386998
// MI455X (gfx1250) — compile-verified
//
#include <hip/hip_runtime.h>

typedef float f32x2 __attribute__((ext_vector_type(2)));
typedef float f32x8 __attribute__((ext_vector_type(8)));

#define IH 768
#define IW 1024
#define DH 384
#define DW 512
#define NBATCH 16
#define TY 24            // DH/16
#define TX 32            // DW/16
#define WPW 4            // waves (tiles) per workgroup
#define PSTRIDE 23       // 22 cols + 1 pad
#define HSTRIDE 17       // 16 cols + 1 pad
#define PT_ELEMS (22 * PSTRIDE)   // 506 floats
#define HF_ELEMS (32 * HSTRIDE)   // 544 floats
#define WAVE_LDS (2 * PT_ELEMS + 5 * HF_ELEMS)  // 3732 floats / wave

// Normalized 7-tap Gaussian, sigma=1.5 (matches reference _gauss_window)
__device__ __forceinline__ float gband(int d) {
    float v = 0.0f;
    v = (d == 0) ? 0.03663285f : v;
    v = (d == 1) ? 0.11128076f : v;
    v = (d == 2) ? 0.21674532f : v;
    v = (d == 3) ? 0.27068214f : v;
    v = (d == 4) ? 0.21674532f : v;
    v = (d == 5) ? 0.11128076f : v;
    v = (d == 6) ? 0.03663285f : v;
    return v;
}

__device__ __forceinline__ f32x8 wmma4(f32x2 a, f32x2 b, f32x8 c) {
    // D(16x16,f32) = A(16x4,f32) x B(4x16,f32) + C
    return __builtin_amdgcn_wmma_f32_16x16x4_f32(
        /*neg_a=*/false, a, /*neg_b=*/false, b,
        /*c_mod=*/(short)0, c, /*reuse_a=*/false, /*reuse_b=*/false);
}

__global__ __launch_bounds__(WPW * 32)
void ssim_main(const float* __restrict__ pred, const float* __restrict__ targ,
               float* __restrict__ acc) {
    __shared__ float lds[WPW * WAVE_LDS];
    __shared__ float red[WPW];

    const int tid  = threadIdx.x;
    const int wave = tid >> 5;
    const int lane = tid & 31;

    float* Pl = lds + wave * WAVE_LDS;
    float* Tl = Pl + PT_ELEMS;
    float* Hl = Tl + PT_ELEMS;

    // ---- tile decode: one 16x16 downsampled output tile per wave ----
    const int tile = blockIdx.x * WPW + wave;
    const int b    = tile / (TY * TX);
    const int rem  = tile - b * (TY * TX);
    const int ty   = rem / TX;
    const int tx   = rem - ty * TX;
    const int R0   = ty * 16;
    const int C0   = tx * 16;

    // ---- phase 1: fused 2x2 avg-pool of 22x22 region into LDS (zero pad) ----
    const float2* p2 = reinterpret_cast<const float2*>(pred);
    const float2* t2 = reinterpret_cast<const float2*>(targ);
    const size_t bbase2 = (size_t)b * (size_t)(IH * (IW / 2));

    for (int i = lane; i < 22 * 22; i += 32) {
        int r  = i / 22, c = i - r * 22;
        int dr = R0 - 3 + r, dc = C0 - 3 + c;
        float pv = 0.0f, tv = 0.0f;
        if (dr >= 0 && dr < DH && dc >= 0 && dc < DW) {
            size_t idx = bbase2 + (size_t)(dr * 2) * (IW / 2) + (size_t)dc;
            float2 pa = p2[idx], pb = p2[idx + IW / 2];
            float2 ta = t2[idx], tb = t2[idx + IW / 2];
            pv = 0.25f * (pa.x + pa.y + pb.x + pb.y);
            tv = 0.25f * (ta.x + ta.y + tb.x + tb.y);
        }
        Pl[r * PSTRIDE + c] = pv;
        Tl[r * PSTRIDE + c] = tv;
    }
    __syncthreads();

    const int lhalf = lane >> 4;   // K-half select in A/B fragments
    const int lrow  = lane & 15;   // M (A) / N (B) index

    // Band-matrix weight fragments: W[j][n] = g[j-n]; identical for the
    // A-side of the vertical pass and B-side of the horizontal pass.
    f32x2 wf[6];
#pragma unroll
    for (int kk = 0; kk < 6; ++kk) {
        int j0 = 4 * kk + 2 * lhalf;
        wf[kk].x = gband(j0 - lrow);
        wf[kk].y = gband(j0 + 1 - lrow);
    }

    // ---- phase 2: horizontal blur via WMMA, 5 fields x 2 row-blocks ----
    // accH order: [p r0, p r1, t r0, t r1, p2 r0, p2 r1, t2 r0, t2 r1, pt r0, pt r1]
    f32x8 accH[10];
#pragma unroll
    for (int i = 0; i < 10; ++i) accH[i] = {};

#pragma unroll
    for (int kk = 0; kk < 6; ++kk) {
        int  j0 = 4 * kk + 2 * lhalf;
        bool c0 = (j0 < 22), c1 = (j0 + 1 < 22);
        int  m0 = lrow, m1 = 16 + lrow;
        bool r1 = (m1 < 22);
        float p00 = c0 ? Pl[m0 * PSTRIDE + j0]     : 0.0f;
        float p01 = c1 ? Pl[m0 * PSTRIDE + j0 + 1] : 0.0f;
        float t00 = c0 ? Tl[m0 * PSTRIDE + j0]     : 0.0f;
        float t01 = c1 ? Tl[m0 * PSTRIDE + j0 + 1] : 0.0f;
        float p10 = (r1 && c0) ? Pl[m1 * PSTRIDE + j0]     : 0.0f;
        float p11 = (r1 && c1) ? Pl[m1 * PSTRIDE + j0 + 1] : 0.0f;
        float t10 = (r1 && c0) ? Tl[m1 * PSTRIDE + j0]     : 0.0f;
        float t11 = (r1 && c1) ? Tl[m1 * PSTRIDE + j0 + 1] : 0.0f;
        f32x2 bw = wf[kk];
        f32x2 a;
        a.x = p00;       a.y = p01;       accH[0] = wmma4(a, bw, accH[0]);
        a.x = p10;       a.y = p11;       accH[1] = wmma4(a, bw, accH[1]);
        a.x = t00;       a.y = t01;       accH[2] = wmma4(a, bw, accH[2]);
        a.x = t10;       a.y = t11;       accH[3] = wmma4(a, bw, accH[3]);
        a.x = p00 * p00; a.y = p01 * p01; accH[4] = wmma4(a, bw, accH[4]);
        a.x = p10 * p10; a.y = p11 * p11; accH[5] = wmma4(a, bw, accH[5]);
        a.x = t00 * t00; a.y = t01 * t01; accH[6] = wmma4(a, bw, accH[6]);
        a.x = t10 * t10; a.y = t11 * t11; accH[7] = wmma4(a, bw, accH[7]);
        a.x = p00 * t00; a.y = p01 * t01; accH[8] = wmma4(a, bw, accH[8]);
        a.x = p10 * t10; a.y = p11 * t11; accH[9] = wmma4(a, bw, accH[9]);
    }

    // ---- store H fields (D-fragment layout: M = v + 8*lhalf, N = lrow) ----
#pragma unroll
    for (int f = 0; f < 5; ++f) {
#pragma unroll
        for (int rb = 0; rb < 2; ++rb) {
            f32x8  av = accH[f * 2 + rb];
            float* Hf = Hl + f * HF_ELEMS;
#pragma unroll
            for (int v = 0; v < 8; ++v) {
                int row = rb * 16 + 8 * lhalf + v;
                Hf[row * HSTRIDE + lrow] = av[v];
            }
        }
    }
    __syncthreads();

    // ---- phase 3: vertical blur via WMMA (weights on A side) ----
    f32x8 accV[5];
#pragma unroll
    for (int f = 0; f < 5; ++f) accV[f] = {};

#pragma unroll
    for (int kk = 0; kk < 6; ++kk) {
        int   j0 = 4 * kk + 2 * lhalf;
        f32x2 aw = wf[kk];
#pragma unroll
        for (int f = 0; f < 5; ++f) {
            const float* Hf = Hl + f * HF_ELEMS;
            f32x2 bv;
            bv.x = Hf[j0 * HSTRIDE + lrow];
            bv.y = Hf[(j0 + 1) * HSTRIDE + lrow];
            accV[f] = wmma4(aw, bv, accV[f]);
        }
    }

    // ---- phase 4: SSIM map + reduction ----
    float s = 0.0f;
#pragma unroll
    for (int v = 0; v < 8; ++v) {
        float mu1 = accV[0][v], mu2 = accV[1][v];
        float bp2 = accV[2][v], bt2 = accV[3][v], bpt = accV[4][v];
        float mu1s = mu1 * mu1, mu2s = mu2 * mu2, mu12 = mu1 * mu2;
        float s1 = bp2 - mu1s, s2 = bt2 - mu2s, s12 = bpt - mu12;
        float num = (2.0f * mu12 + 1e-4f) * (2.0f * s12 + 9e-4f);
        float den = (mu1s + mu2s + 1e-4f) * (s1 + s2 + 9e-4f);
        s += num / (den + 1e-12f);
    }

#pragma unroll
    for (int off = 16; off > 0; off >>= 1)
        s += __shfl_down(s, off, 32);
    if (lane == 0) red[wave] = s;
    __syncthreads();
    if (tid == 0) {
        float tot = red[0];
        for (int w = 1; w < WPW; ++w) tot += red[w];
        atomicAdd(acc, tot);
    }
}

__global__ void ssim_init(float* acc) { acc[0] = 0.0f; }

__global__ void ssim_fin(const float* __restrict__ acc, float* __restrict__ out) {
    out[0] = 1.0f - acc[0] * (1.0f / (float)(NBATCH * DH * DW));
}

extern "C" void kernel_launch(void* const* d_in, const int* in_sizes, int n_in,
                              void* d_out, int out_size, void* d_ws, size_t ws_size,
                              hipStream_t stream) {
    (void)in_sizes; (void)n_in; (void)out_size; (void)ws_size;
    const float* pred = (const float*)d_in[0];
    const float* targ = (const float*)d_in[1];
    float* out = (float*)d_out;
    float* acc = (float*)d_ws;

    ssim_init<<<1, 1, 0, stream>>>(acc);
    const int nblocks = (NBATCH * TY * TX) / WPW;  // 12288 tiles / 4 = 3072
    ssim_main<<<nblocks, WPW * 32, 0, stream>>>(pred, targ, acc);
    ssim_fin<<<1, 1, 0, stream>>>(acc, out);
}